// NodeModel_13108240188139
// MI455X (gfx1250) — compile-verified
//
#include <hip/hip_runtime.h>
#include <hip/hip_bf16.h>

typedef __attribute__((ext_vector_type(2))) float v2f;
typedef __attribute__((ext_vector_type(8))) float v8f;

#define DFEAT 128

// ---------------------------------------------------------------- utilities
__global__ __launch_bounds__(256) void zero_f32(float* __restrict__ p, int n) {
    int i = blockIdx.x * 256 + threadIdx.x;
    if (i < n) p[i] = 0.0f;
}

// One wave per edge row: lane handles 4 contiguous floats, 4 f32 atomics to L2.
__global__ __launch_bounds__(256) void scatter_add_kernel(
    const float* __restrict__ ea, const int* __restrict__ src,
    float* __restrict__ agg, int nE)
{
    int e = blockIdx.x * 8 + (threadIdx.x >> 5);
    if (e >= nE) return;
    int lane = threadIdx.x & 31;
    int s = src[e];
    const float4 v = *(const float4*)(ea + (size_t)e * DFEAT + lane * 4);
    float* dst = agg + (size_t)s * DFEAT + lane * 4;
    atomicAdd(dst + 0, v.x);
    atomicAdd(dst + 1, v.y);
    atomicAdd(dst + 2, v.z);
    atomicAdd(dst + 3, v.w);
}

// stats[0..127]=colsum, stats[128..255]=colsumsq  ->  affine a,c (BN folded)
__global__ void finalize_bn(const float* __restrict__ stats,
                            const float* __restrict__ gamma,
                            const float* __restrict__ beta,
                            float* __restrict__ affine, float invN)
{
    int j = threadIdx.x;            // 128 threads
    float mean = stats[j] * invN;
    float var  = stats[DFEAT + j] * invN - mean * mean;
    float a = rsqrtf(var + 1e-5f) * gamma[j];
    affine[j]         = a;
    affine[DFEAT + j] = beta[j] - mean * a;
}

// ---------------------------------------------------------------- WMMA GEMM
// out[N,128] = act( A[N,KDIM] @ W[KDIM,128] + bias ), persistent grid.
// CONCAT: A = concat(A0, A1) along K (KDIM=256).
// AFFINE: element A[r,k] replaced by A*affine[k] + affine[128+k] (BN fold).
// SILU:   SiLU activation. STATS: accumulate column sum/sumsq into stats.
template<int KDIM, bool CONCAT, bool AFFINE, bool SILU, bool STATS>
__global__ __launch_bounds__(256) void gemm_kernel(
    const float* __restrict__ A0, const float* __restrict__ A1,
    const float* __restrict__ affine,
    const float* __restrict__ W, const float* __restrict__ bias,
    float* __restrict__ out, float* __restrict__ stats,
    int nTiles, int nRows)
{
    constexpr int KP = KDIM + 2;      // +2 floats: conflict-free b64 LDS reads
    extern __shared__ float lds[];
    float* Wt   = lds;                 // [128][KP]  W transposed
    float* At   = lds + 128 * KP;      // [16][KP]   A tile
    float* sred = At + 16 * KP;        // [256]      stats reduction

    const int tid   = threadIdx.x;
    const int wave  = tid >> 5;
    const int lane  = tid & 31;
    const int lhalf = lane & 15;
    const int khalf = (lane >> 4) << 1;   // 0 | 2 : K sub-offset per half-wave
    const int rhi   = (lane >> 4) << 3;   // 0 | 8 : M offset of C rows
    const int col   = wave * 16 + lhalf;  // this lane's output column

    // Load W transposed into LDS (once per block, amortized by grid-stride).
    for (int idx = tid; idx < KDIM * DFEAT; idx += 256) {
        int k = idx >> 7;          // idx / 128
        int n = idx & (DFEAT - 1); // idx % 128
        Wt[n * KP + k] = W[idx];
    }

    const float bcol = bias[col];
    float s1 = 0.0f, s2 = 0.0f;

    for (int t = blockIdx.x; t < nTiles; t += gridDim.x) {
        __syncthreads();               // At reuse across iterations + Wt ready
        const int rowBase = t * 16;

        // ---- load 16 x KDIM A tile (float4), fusing concat / BN affine ----
        constexpr int NF4 = 16 * KDIM / 4;
        for (int f = tid; f < NF4; f += 256) {
            int r  = f / (KDIM / 4);
            int k  = (f % (KDIM / 4)) * 4;
            int row = rowBase + r;
            float4 v = make_float4(0.f, 0.f, 0.f, 0.f);
            if (row < nRows) {
                if (CONCAT) {
                    const float* sp = (k < DFEAT)
                        ? (A0 + (size_t)row * DFEAT + k)
                        : (A1 + (size_t)row * DFEAT + (k - DFEAT));
                    v = *(const float4*)sp;
                } else {
                    v = *(const float4*)(A0 + (size_t)row * KDIM + k);
                }
            }
            if (AFFINE) {
                float4 a = *(const float4*)(affine + k);
                float4 c = *(const float4*)(affine + DFEAT + k);
                v.x = v.x * a.x + c.x;  v.y = v.y * a.y + c.y;
                v.z = v.z * a.z + c.z;  v.w = v.w * a.w + c.w;
            }
            float* dp = At + r * KP + k;
            dp[0] = v.x; dp[1] = v.y; dp[2] = v.z; dp[3] = v.w;
        }
        __syncthreads();

        // ---- K loop: V_WMMA_F32_16X16X4_F32 ----
        v8f acc = {0.f, 0.f, 0.f, 0.f, 0.f, 0.f, 0.f, 0.f};
        #pragma unroll
        for (int k0 = 0; k0 < KDIM; k0 += 4) {
            // A frag: lane<16 -> A[m][k0],A[m][k0+1]; lane>=16 -> k0+2,k0+3
            const float* ap = At + lhalf * KP + k0 + khalf;
            v2f a; a[0] = ap[0]; a[1] = ap[1];
            // B frag from transposed W: same K split, column = this lane's n
            const float* bp = Wt + col * KP + k0 + khalf;
            v2f b; b[0] = bp[0]; b[1] = bp[1];
            acc = __builtin_amdgcn_wmma_f32_16x16x4_f32(
                false, a, false, b, (short)0, acc, false, false);
        }

        // ---- epilogue: bias (+SiLU) store, per-lane column stats ----
        #pragma unroll
        for (int i = 0; i < 8; ++i) {
            int row = rowBase + rhi + i;
            if (row < nRows) {
                float v = acc[i] + bcol;
                if (SILU) v = v / (1.0f + __expf(-v));
                out[(size_t)row * DFEAT + col] = v;
                if (STATS) { s1 += v; s2 += v * v; }
            }
        }
    }

    if (STATS) {
        __syncthreads();
        sred[tid] = 0.0f;              // tid < 256 always
        __syncthreads();
        atomicAdd(&sred[col], s1);             // ds_add_f32
        atomicAdd(&sred[DFEAT + col], s2);
        __syncthreads();
        atomicAdd(&stats[tid], sred[tid]);     // one global atomic per thread
    }
}

// ---------------------------------------------------------------- launcher
extern "C" void kernel_launch(void* const* d_in, const int* in_sizes, int n_in,
                              void* d_out, int out_size, void* d_ws, size_t ws_size,
                              hipStream_t stream) {
    const float* x   = (const float*)d_in[0];
    const int*   ei  = (const int*)  d_in[1];
    const float* ea  = (const float*)d_in[2];
    const float* W1  = (const float*)d_in[3];
    const float* b1  = (const float*)d_in[4];
    const float* g1  = (const float*)d_in[5];
    const float* be1 = (const float*)d_in[6];
    const float* W2  = (const float*)d_in[7];
    const float* b2  = (const float*)d_in[8];
    const float* g2  = (const float*)d_in[9];
    const float* be2 = (const float*)d_in[10];
    const float* W3  = (const float*)d_in[11];
    const float* b3  = (const float*)d_in[12];
    float* out = (float*)d_out;

    const int N  = in_sizes[0] / DFEAT;   // 50000
    const int nE = in_sizes[2] / DFEAT;   // 600000
    const int* src = ei;                  // edge_index[0] = first nE ints

    float* ws    = (float*)d_ws;
    float* agg   = ws;
    float* h1    = agg + (size_t)N * DFEAT;
    float* h2    = h1  + (size_t)N * DFEAT;
    float* stats = h2  + (size_t)N * DFEAT;  // [512]: stats1, stats2
    float* aff1  = stats + 512;              // [256]
    float* aff2  = aff1 + 256;               // [256]

    // zero agg + stats
    {
        int n = N * DFEAT;
        zero_f32<<<(n + 255) / 256, 256, 0, stream>>>(agg, n);
        zero_f32<<<2, 256, 0, stream>>>(stats, 512);
    }

    // scatter-add edge features by source node
    scatter_add_kernel<<<(nE + 7) / 8, 256, 0, stream>>>(ea, src, agg, nE);

    const int nTiles = (N + 15) / 16;        // 3125
    const int grid   = nTiles < 640 ? nTiles : 640;
    const size_t lds1 = (size_t)(128 * (256 + 2) + 16 * (256 + 2) + 256) * sizeof(float);
    const size_t lds2 = (size_t)(128 * (128 + 2) + 16 * (128 + 2) + 256) * sizeof(float);
    const float invN = 1.0f / (float)N;

    // layer 1: h1 = SiLU(concat(x, agg) @ W1 + b1), accumulate stats1
    gemm_kernel<256, true, false, true, true><<<grid, 256, lds1, stream>>>(
        x, agg, nullptr, W1, b1, h1, stats, nTiles, N);
    finalize_bn<<<1, 128, 0, stream>>>(stats, g1, be1, aff1, invN);

    // layer 2: h2 = SiLU(BN(h1) @ W2 + b2), accumulate stats2
    gemm_kernel<128, false, true, true, true><<<grid, 256, lds2, stream>>>(
        h1, nullptr, aff1, W2, b2, h2, stats + 256, nTiles, N);
    finalize_bn<<<1, 128, 0, stream>>>(stats + 256, g2, be2, aff2, invN);

    // layer 3: out = BN(h2) @ W3 + b3  (no activation, no stats)
    gemm_kernel<128, false, true, false, false><<<grid, 256, lds2, stream>>>(
        h2, nullptr, aff2, W3, b3, out, nullptr, nTiles, N);
}